// Quan_30992484008182
// MI455X (gfx1250) — compile-verified
//
#include <hip/hip_runtime.h>
#include <cstdint>

#define TPB   256   // 8 waves (wave32)
#define TILES 8     // tiles of 256 float4 per block, double-buffered

// Closed-form staircase: only the nearest integer offset contributes a
// non-saturated sigmoid (R=100 => deviation <= e^-50 elsewhere).
__device__ __forceinline__ float my_ceil_elem(float xv) {
    float m  = rintf(xv);                       // v_rndne_f32
    float z  = 100.0f * (xv - m);               // |z| <= 50
    float ez = __expf(-fabsf(z));               // v_exp_f32 path
    float p  = __builtin_amdgcn_rcpf(1.0f + ez);// sigma(|z|)
    float s  = (z >= 0.0f) ? p : (1.0f - p);    // sigma(z), stable
    float am = fabsf(m);
    float fr = (am >= 1.0f && am <= 150.0f) ? s : 0.0f;
    float bs = fminf(fmaxf(m, -150.0f), 0.0f)
             + fminf(fmaxf(m - 1.0f, 0.0f), 150.0f);
    return bs + fr;
}

__global__ __launch_bounds__(TPB) void quan_stair_main(
    const float* __restrict__ x, const float* __restrict__ kern,
    float* __restrict__ out)
{
    __shared__ float4 buf[2][TPB];
    __shared__ float  skern[12];

    const unsigned tid = threadIdx.x;
    if (tid < 12) skern[tid] = kern[tid];
    __syncthreads();

    const unsigned i4_0 = (blockIdx.x * TILES) * TPB + tid;   // float4 index
    unsigned long long gaddr = (unsigned long long)(uintptr_t)x
                             + (unsigned long long)i4_0 * 16ull;
    // Generic shared pointers carry the LDS byte offset in the low 32 bits.
    unsigned lds0 = (unsigned)(uintptr_t)&buf[0][tid];
    unsigned lds1 = (unsigned)(uintptr_t)&buf[1][tid];

    // Prime the pipeline: async copy tile 0 into LDS (ASYNCcnt-tracked).
    asm volatile("global_load_async_to_lds_b128 %0, %1, off"
                 :: "v"(lds0), "v"(gaddr) : "memory");

    float4* out4 = (float4*)out;

    #pragma unroll
    for (int t = 0; t < TILES; ++t) {
        if (t + 1 < TILES) {
            unsigned long long gnext =
                gaddr + (unsigned long long)(t + 1) * (TPB * 16ull);
            asm volatile("global_load_async_to_lds_b128 %0, %1, off"
                         :: "v"(((t + 1) & 1) ? lds1 : lds0), "v"(gnext)
                         : "memory");
            // keep the just-issued load in flight, wait for tile t only
            asm volatile("s_wait_asynccnt 0x1" ::: "memory");
        } else {
            asm volatile("s_wait_asynccnt 0x0" ::: "memory");
        }

        float4 v = buf[t & 1][tid];                 // ds_load_b128

        unsigned i4 = i4_0 + (unsigned)t * TPB;
        // 4 consecutive floats starting at a multiple of 4 => channels
        // {0..3},{4..7},{8..11}; never wraps mod 12.
        unsigned c0 = 4u * (i4 % 3u);
        float k0 = skern[c0 + 0], k1 = skern[c0 + 1],
              k2 = skern[c0 + 2], k3 = skern[c0 + 3];

        float4 o;
        o.x = my_ceil_elem(v.x * k0);
        o.y = my_ceil_elem(v.y * k1);
        o.z = my_ceil_elem(v.z * k2);
        o.w = my_ceil_elem(v.w * k3);
        out4[i4] = o;                               // global_store_b128
    }
}

__global__ __launch_bounds__(TPB) void quan_stair_tail(
    const float* __restrict__ x, const float* __restrict__ kern,
    float* __restrict__ out, long long start, long long n)
{
    long long i = start + (long long)blockIdx.x * TPB + threadIdx.x;
    if (i < n) {
        unsigned c = (unsigned)((unsigned long long)i % 12ull);
        out[i] = my_ceil_elem(x[i] * kern[c]);
    }
}

extern "C" void kernel_launch(void* const* d_in, const int* in_sizes, int n_in,
                              void* d_out, int out_size, void* d_ws, size_t ws_size,
                              hipStream_t stream) {
    const float* x = (const float*)d_in[0];
    const float* k = (const float*)d_in[1];
    float* out = (float*)d_out;

    const long long n = (long long)in_sizes[0];              // 25,165,824 here
    const long long elemsPerBlock = (long long)TPB * 4 * TILES; // 8192
    const long long nMain = (n / elemsPerBlock) * elemsPerBlock;
    const int blocks = (int)(nMain / elemsPerBlock);          // 3072 here

    if (blocks > 0)
        quan_stair_main<<<blocks, TPB, 0, stream>>>(x, k, out);

    if (nMain < n) {
        long long rem = n - nMain;
        int tb = (int)((rem + TPB - 1) / TPB);
        quan_stair_tail<<<tb, TPB, 0, stream>>>(x, k, out, nMain, n);
    }
}